// OrdinalLoss_89678917141332
// MI455X (gfx1250) — compile-verified
//
#include <hip/hip_runtime.h>

// OrdinalLoss on MI455X (gfx1250, wave32).
//
// error = sum_{i,c} contrib(i,c) * mask(i) / (b * n_t),  a single scalar.
//
// Per element (thresholds sorted, target slot j of channel c):
//   contrib = G(p) - (n_j*p - P_j),   G(p) = sum_k relu(p - thr_k)
// with P_j = prefix sum of thresholds, n_j = j+1; (0,0) for non-targets.
// Derivation: relu(x) - relu(-x) = x applied to the m_k = +1 subset.
// (P_j, n_j) pairs are scattered into a per-wave LDS table (DS pipe) and
// un-scattered (re-zeroed) after each row, so the table is zeroed only once.
// VALU cost ~12 slots/element (dual-issued v_max + v_pk_* VOP3P).
//
// Accumulation AND the per-row mask multiply run on the matrix pipe via
// V_WMMA_F32_16X16X4_F32: D += A x B with A = mask[row] * ones(16x4) and
// B = fresh contributions, so D's rows hold running masked column sums.
// Memory floor ~0.75us (17.3MB @ 23.3TB/s); VALU is the binding pipe.

typedef __attribute__((ext_vector_type(2))) float v2f;
typedef __attribute__((ext_vector_type(8))) float v8f;

#define B_DIM 32
#define L_DIM 512
#define C_DIM 256
#define NT 8
#define N_ROWS (B_DIM * L_DIM)                 // 16384
#define BLOCKS 512
#define THREADS 256                             // 8 waves of 32 on CDNA5
#define WPB (THREADS / 32)
#define TOTAL_WAVES (BLOCKS * WPB)              // 4096
#define ROWS_PER_WAVE (N_ROWS / TOTAL_WAVES)    // 4

__device__ __forceinline__ void wave_fence() {
    // Compiler memory fence + wave-synchronous scheduling barrier: LDS ops are
    // in-order per wave at ISA level; this stops the compiler reordering the
    // cross-lane-dependent DS stores/loads around each other.
    asm volatile("" ::: "memory");
    __builtin_amdgcn_wave_barrier();
}

__device__ __forceinline__ float contrib(float pv, float P, float nf,
                                         const float thr[NT]) {
    float g = 0.0f;
#pragma unroll
    for (int k = 0; k < NT; ++k)
        g += fmaxf(pv - thr[k], 0.0f);      // G(p): dual-issue max, pk adds
    return fmaf(-nf, pv, P) + g;            // G - (n*p - P)
}

__global__ __launch_bounds__(THREADS)
void OrdinalLoss_partial_kernel(const float* __restrict__ pred,
                                const float* __restrict__ thr_g,
                                const int*   __restrict__ tgt_g,
                                const float* __restrict__ mask,
                                float* __restrict__ partial) {
    const int lane  = threadIdx.x & 31;
    const int wave  = threadIdx.x >> 5;
    const int gwave = blockIdx.x * WPB + wave;

    // Thresholds: wave-uniform -> SGPRs.
    float thr[NT];
#pragma unroll
    for (int k = 0; k < NT; ++k) thr[k] = thr_g[k];

    // Prefix sums: lane j (j<8) keeps (P_j, j+1) for the scatter.
    float Pacc = 0.0f, myP = 0.0f, myN = 0.0f;
#pragma unroll
    for (int j = 0; j < NT; ++j) {
        Pacc += thr[j];
        if (lane == j) { myP = Pacc; myN = (float)(j + 1); }
    }

    // Per-wave (P, n) table: C_DIM float2 entries (2KB/wave). Zeroed ONCE;
    // per row we scatter 8 entries and un-scatter them afterwards.
    __shared__ float tmshared[WPB * C_DIM * 2];
    float* wbuf = tmshared + wave * C_DIM * 2;
    {
        float4* wz = (float4*)wbuf;
#pragma unroll
        for (int t = 0; t < 4; ++t)
            wz[lane + t * 32] = make_float4(0.f, 0.f, 0.f, 0.f);
    }
    wave_fence();

    v8f acc = {};                               // WMMA C/D accumulator

    for (int rr = 0; rr < ROWS_PER_WAVE; ++rr) {
        const int row   = gwave + rr * TOTAL_WAVES;
        const float rmw = mask[row];
        const float* __restrict__ prow = pred + (size_t)row * C_DIM;
        const v2f maskA = {rmw, rmw};           // A = rmw*ones: mask mul on matrix pipe

        // Prefetch next strided row into GL2 (speculative; OOB dropped).
        if (rr + 1 < ROWS_PER_WAVE)
            __builtin_prefetch(prow + (size_t)TOTAL_WAVES * C_DIM + lane * 4, 0, 0);

        // One target index per lane (lanes 0..7 scatter).
        const int tval = tgt_g[row * NT + (lane & (NT - 1))];

        if (lane < NT) ((float2*)wbuf)[tval] = make_float2(myP, myN);
        wave_fence();

#pragma unroll
        for (int q = 0; q < 2; ++q) {
            const int c0 = q * 128 + lane * 4;           // coalesced 512B/wave
            const float4 p4 = *(const float4*)(prow + c0);
            const float4 Qa = *(const float4*)(wbuf + c0 * 2);      // (P0,n0,P1,n1)
            const float4 Qb = *(const float4*)(wbuf + c0 * 2 + 4);  // (P2,n2,P3,n3)

            v2f b0, b1;
            b0[0] = contrib(p4.x, Qa.x, Qa.y, thr);
            b0[1] = contrib(p4.y, Qa.z, Qa.w, thr);
            b1[0] = contrib(p4.z, Qb.x, Qb.y, thr);
            b1[1] = contrib(p4.w, Qb.z, Qb.w, thr);

            // Matrix-pipe accumulate, mask folded into A (EXEC all-1s here).
            acc = __builtin_amdgcn_wmma_f32_16x16x4_f32(
                false, maskA, false, b0, (short)0, acc, false, false);
            acc = __builtin_amdgcn_wmma_f32_16x16x4_f32(
                false, maskA, false, b1, (short)0, acc, false, false);
        }
        wave_fence();   // loads of this row complete-order before un-scatter

        // Un-scatter: restore the 8 touched entries to zero for the next row.
        if (lane < NT) ((float2*)wbuf)[tval] = make_float2(0.f, 0.f);
        wave_fence();
    }

    // D rows identical; D vgpr0 lanes 0..15 = the 16 masked column sums.
    __shared__ float red[WPB * 16];
    if (lane < 16) red[wave * 16 + lane] = acc[0];
    __syncthreads();

#pragma unroll
    for (int off = 64; off >= 1; off >>= 1) {
        if (threadIdx.x < off) red[threadIdx.x] += red[threadIdx.x + off];
        __syncthreads();
    }
    if (threadIdx.x == 0) partial[blockIdx.x] = red[0];
}

__global__ __launch_bounds__(256)
void OrdinalLoss_final_reduce(const float* __restrict__ partial,
                              float* __restrict__ out, int n, float scale) {
    __shared__ float s[256];
    float v = 0.0f;
    for (int i = threadIdx.x; i < n; i += 256) v += partial[i];
    s[threadIdx.x] = v;
    __syncthreads();
#pragma unroll
    for (int off = 128; off >= 1; off >>= 1) {
        if (threadIdx.x < off) s[threadIdx.x] += s[threadIdx.x + off];
        __syncthreads();
    }
    if (threadIdx.x == 0) out[0] = s[0] * scale;
}

extern "C" void kernel_launch(void* const* d_in, const int* in_sizes, int n_in,
                              void* d_out, int out_size, void* d_ws, size_t ws_size,
                              hipStream_t stream) {
    const float* pred = (const float*)d_in[0];   // (32,512,256) f32
    const float* thr  = (const float*)d_in[1];   // (8,) f32
    const int*   tgt  = (const int*)  d_in[2];   // (32,512,8) i32
    const float* mask = (const float*)d_in[3];   // (32,512) f32
    float* out     = (float*)d_out;              // scalar f32
    float* partial = (float*)d_ws;               // BLOCKS floats of scratch

    OrdinalLoss_partial_kernel<<<BLOCKS, THREADS, 0, stream>>>(
        pred, thr, tgt, mask, partial);

    const float scale = 1.0f / (float)(B_DIM * NT);   // mean over b, / n_t
    OrdinalLoss_final_reduce<<<1, 256, 0, stream>>>(partial, out, BLOCKS, scale);
}